// GraphInputProcessorHet_16432544874896
// MI455X (gfx1250) — compile-verified
//
#include <hip/hip_runtime.h>
#include <hip/hip_bf16.h>

// Problem constants (from reference): B=512 batch, N=128 nodes, F=N+1=129, D=256.
#define NB   512
#define NN   128
#define NF   129
#define ND   256

typedef __attribute__((ext_vector_type(2))) float v2f;
typedef __attribute__((ext_vector_type(4))) float v4f;
typedef __attribute__((ext_vector_type(8))) float v8f;

// tanh via hardware transcendentals: tanh(x) = 1 - 2 * rcp(exp(2x) + 1).
// __expf -> v_exp_f32, __builtin_amdgcn_rcpf -> v_rcp_f32 (no IEEE div chain).
// Saturates correctly at +/-1 for |x| large (exp -> inf / 0).
__device__ __forceinline__ float fast_tanh(float x) {
    float e = __expf(2.0f * x);
    return 1.0f - 2.0f * __builtin_amdgcn_rcpf(e + 1.0f);
}

// h[b,n,d] = tanh(z[b,n]*W[n,0,d] + W[n,1+n,d] + bias[n,d])
// One WMMA_F32_16X16X4_F32 per (16-batch x 16-d) tile:
//   A[16x4]: column K=0 = z[b0..b0+15, n], rest 0
//   B[4x16]: row    K=0 = W[n,0,d0..d0+15], rest 0
//   C[16x16]: every row = W[n,1+n,d0+j] + bias[n,d0+j]  (broadcast over M)
__global__ void __launch_bounds__(256)
het_rank1_tanh_wmma(const float* __restrict__ z,
                    const float* __restrict__ W,
                    const float* __restrict__ bias,
                    float* __restrict__ h)
{
    const int n    = blockIdx.x;        // node 0..127
    const int b0   = blockIdx.y << 4;   // batch tile base
    const int lane = threadIdx.x & 31;
    const int wave = threadIdx.x >> 5;  // 0..7, each wave owns 2 d-tiles
    const int l16  = lane & 15;
    const int hi   = lane >> 4;         // 0: lanes 0-15, 1: lanes 16-31

    // A operand (same for both d-tiles of this wave)
    const float zv = z[(size_t)(b0 + l16) * NN + n];
    v2f a;
    a.x = hi ? 0.0f : zv;   // VGPR0: K=0 (lanes 0-15), K=2 (lanes 16-31, zero)
    a.y = 0.0f;             // VGPR1: K=1 / K=3 -> zero

    const size_t wrow0 = (size_t)n * NF * ND;                   // W[n,0,:]
    const size_t wrowd = wrow0 + (size_t)(n + 1) * ND;          // W[n,1+n,:]
    const size_t brow  = (size_t)n * ND;                        // bias[n,:]

#pragma unroll
    for (int t = 0; t < 2; ++t) {
        const int d0 = ((wave << 1) + t) << 4;                  // d-tile base

        const float w0 = W[wrow0 + d0 + l16];
        v2f bm;
        bm.x = hi ? 0.0f : w0;  // row K=0 of B in lanes 0-15
        bm.y = 0.0f;

        const float cv = W[wrowd + d0 + l16] + bias[brow + d0 + l16];
        v8f c;
#pragma unroll
        for (int i = 0; i < 8; ++i) c[i] = cv;   // bias broadcast over all M rows

        // (neg_a, A, neg_b, B, c_mod, C, reuse_a, reuse_b)
        c = __builtin_amdgcn_wmma_f32_16x16x4_f32(
                false, a, false, bm, (short)0, c, false, false);

        // D layout: VGPR r -> M = r + 8*hi, N = l16
#pragma unroll
        for (int r = 0; r < 8; ++r) {
            const int m = r + (hi << 3);
            __builtin_nontemporal_store(
                fast_tanh(c[r]),
                &h[((size_t)(b0 + m) * NN + n) * ND + d0 + l16]);
        }
    }
}

// Streams edge_index (row/col), edge_weights, edge_types.
// e = b*N*N + r*N + c ; row = b*N + r ; col = b*N + c ;
// weight = adj[r,c] ; type = r*N + c + 1.
// One thread per 4 consecutive edges -> 16B non-temporal stores.
__global__ void __launch_bounds__(256)
edges_stream(const float* __restrict__ adj, float* __restrict__ out)
{
    const size_t BNN  = (size_t)NB * NN * NN;       // 8388608
    const size_t H_SZ = (size_t)NB * NN * ND;       // 16777216
    const size_t EI   = H_SZ;                       // rows
    const size_t EC   = EI + BNN;                   // cols
    const size_t EW   = EI + 2 * BNN;               // weights
    const size_t ET   = EW + BNN;                   // types

    const size_t e4 = ((size_t)blockIdx.x * blockDim.x + threadIdx.x) << 2;
    if (e4 >= BNN) return;

    const int b  = (int)(e4 >> 14);      // / (N*N)
    const int rc = (int)(e4 & 16383);    // % (N*N)
    const int r  = rc >> 7;

    const float rowv = (float)(b * NN + r);
    const float col0 = (float)(b * NN + (rc & 127));
    const float ty0  = (float)(rc + 1);

    const v4f wq   = *reinterpret_cast<const v4f*>(adj + rc);   // 16B aligned
    const v4f rowq = {rowv, rowv, rowv, rowv};
    const v4f colq = {col0, col0 + 1.0f, col0 + 2.0f, col0 + 3.0f};
    const v4f tyq  = {ty0,  ty0 + 1.0f,  ty0 + 2.0f,  ty0 + 3.0f};

    __builtin_nontemporal_store(rowq, reinterpret_cast<v4f*>(out + EI + e4));
    __builtin_nontemporal_store(colq, reinterpret_cast<v4f*>(out + EC + e4));
    __builtin_nontemporal_store(wq,   reinterpret_cast<v4f*>(out + EW + e4));
    __builtin_nontemporal_store(tyq,  reinterpret_cast<v4f*>(out + ET + e4));
}

extern "C" void kernel_launch(void* const* d_in, const int* in_sizes, int n_in,
                              void* d_out, int out_size, void* d_ws, size_t ws_size,
                              hipStream_t stream) {
    (void)in_sizes; (void)n_in; (void)out_size; (void)d_ws; (void)ws_size;

    const float* z    = (const float*)d_in[0];  // [B, N]
    const float* adj  = (const float*)d_in[1];  // [N, N]
    const float* W    = (const float*)d_in[2];  // [N, F, D]
    const float* bias = (const float*)d_in[3];  // [N, D]
    float* out = (float*)d_out;

    // h: one block per (node, 16-batch tile); 8 waves x 2 d-tiles = D=256
    dim3 gridH(NN, NB / 16);
    het_rank1_tanh_wmma<<<gridH, 256, 0, stream>>>(z, W, bias, out);

    // edges: BNN/4 threads
    const size_t BNN = (size_t)NB * NN * NN;
    const int blocks = (int)((BNN / 4 + 255) / 256);   // 8192
    edges_stream<<<blocks, 256, 0, stream>>>(adj, out);
}